// GraphSAGEwithEmbed_23381801959789
// MI455X (gfx1250) — compile-verified
//
#include <hip/hip_runtime.h>

typedef __attribute__((ext_vector_type(2))) float v2f;
typedef __attribute__((ext_vector_type(8))) float v8f;

#define NFEAT 128

__device__ __forceinline__ v8f wmma_f32_k4(v2f a, v2f b, v8f c) {
  // V_WMMA_F32_16X16X4_F32: D = A(16x4) * B(4x16) + C, all fp32
  return __builtin_amdgcn_wmma_f32_16x16x4_f32(false, a, false, b, (short)0, c,
                                               false, false);
}

// D[M,N] = A[M,K] @ W[N,K]^T (+bias) (+relu).  N == 128, block = 256 (8 waves),
// each wave computes one 16x16 tile; grid.x covers M in 16-row tiles.
__global__ void gemm_nt_wmma(const float* __restrict__ A,
                             const float* __restrict__ W,
                             const float* __restrict__ bias,
                             float* __restrict__ D,
                             int M, int N, int K, int relu) {
  const int lane  = threadIdx.x & 31;
  const int wave  = threadIdx.x >> 5;
  const int nBase = wave * 16;
  if (nBase >= N) return;
  const int mBase = blockIdx.x * 16;
  const int r = lane & 15;   // A-row / B-col within tile
  const int h = lane >> 4;   // K half-select
  int ar = mBase + r; if (ar > M - 1) ar = M - 1;   // clamp (EXEC stays full)
  const float* Arow = A + (size_t)ar * K;
  const float* Wrow = W + (size_t)(nBase + r) * K;
  v8f c = {0.f, 0.f, 0.f, 0.f, 0.f, 0.f, 0.f, 0.f};
  for (int k = 0; k < K; k += 4) {
    v2f a = *(const v2f*)(Arow + k + 2 * h);
    v2f b = *(const v2f*)(Wrow + k + 2 * h);
    c = wmma_f32_k4(a, b, c);
  }
  const int col = nBase + r;
  const float bv = bias ? bias[col] : 0.f;
#pragma unroll
  for (int i = 0; i < 8; ++i) {
    int row = mBase + i + 8 * h;
    if (row < M) {
      float v = c[i] + bv;
      if (relu) v = fmaxf(v, 0.f);
      D[(size_t)row * N + col] = v;
    }
  }
}

// Edge scatter: one wave per edge; 32 lanes x 4 feats = 128. Lane 0 counts degree.
__global__ void sage_scatter(const float* __restrict__ H,
                             const int* __restrict__ src,
                             const int* __restrict__ dst,
                             float* __restrict__ Agg,
                             float* __restrict__ Deg,
                             int nEdges) {
  unsigned long long gid =
      (unsigned long long)blockIdx.x * blockDim.x + threadIdx.x;
  unsigned e = (unsigned)(gid >> 5);
  if (e >= (unsigned)nEdges) return;
  int t = (int)(gid & 31);
  int s = src[e], d = dst[e];
  const float* hs = H + (size_t)s * NFEAT + t * 4;
  float* ag = Agg + (size_t)d * NFEAT + t * 4;
  atomicAdd(ag + 0, hs[0]);
  atomicAdd(ag + 1, hs[1]);
  atomicAdd(ag + 2, hs[2]);
  atomicAdd(ag + 3, hs[3]);
  if (t == 0) atomicAdd(Deg + d, 1.0f);
}

// D = Hprev[:Mdst] @ Wself^T + (Agg/max(Deg,1)) @ Wneigh^T + b  (+relu)
// K = N = 128 fixed.
__global__ void sage_combine_wmma(const float* __restrict__ Hprev,
                                  const float* __restrict__ Agg,
                                  const float* __restrict__ Deg,
                                  const float* __restrict__ Wself,
                                  const float* __restrict__ Wneigh,
                                  const float* __restrict__ bias,
                                  float* __restrict__ D,
                                  int Mdst, int relu) {
  const int K = NFEAT;
  const int lane  = threadIdx.x & 31;
  const int wave  = threadIdx.x >> 5;
  const int nBase = wave * 16;
  const int mBase = blockIdx.x * 16;
  const int r = lane & 15;
  const int h = lane >> 4;
  int ar = mBase + r; if (ar > Mdst - 1) ar = Mdst - 1;
  const float* Hrow = Hprev + (size_t)ar * K;
  const float* Grow = Agg   + (size_t)ar * K;
  const float  inv  = 1.0f / fmaxf(Deg[ar], 1.0f);
  const float* WsR  = Wself  + (size_t)(nBase + r) * K;
  const float* WnR  = Wneigh + (size_t)(nBase + r) * K;
  v8f c = {0.f, 0.f, 0.f, 0.f, 0.f, 0.f, 0.f, 0.f};
#pragma unroll 4
  for (int k = 0; k < K; k += 4) {
    v2f a = *(const v2f*)(Hrow + k + 2 * h);
    v2f b = *(const v2f*)(WsR + k + 2 * h);
    c = wmma_f32_k4(a, b, c);
  }
#pragma unroll 4
  for (int k = 0; k < K; k += 4) {
    v2f g = *(const v2f*)(Grow + k + 2 * h);
    v2f a; a.x = g.x * inv; a.y = g.y * inv;
    v2f b = *(const v2f*)(WnR + k + 2 * h);
    c = wmma_f32_k4(a, b, c);
  }
  const int col = nBase + r;
  const float bv = bias[col];
#pragma unroll
  for (int i = 0; i < 8; ++i) {
    int row = mBase + i + 8 * h;
    if (row < Mdst) {
      float v = c[i] + bv;
      if (relu) v = fmaxf(v, 0.f);
      D[(size_t)row * NFEAT + col] = v;
    }
  }
}

extern "C" void kernel_launch(void* const* d_in, const int* in_sizes, int n_in,
                              void* d_out, int out_size, void* d_ws,
                              size_t ws_size, hipStream_t stream) {
  (void)in_sizes; (void)n_in; (void)out_size; (void)ws_size;
  const float* x      = (const float*)d_in[0];
  const int*   src0   = (const int*)d_in[1];
  const int*   dst0   = (const int*)d_in[2];
  const int*   src1   = (const int*)d_in[3];
  const int*   dst1   = (const int*)d_in[4];
  const int*   src2   = (const int*)d_in[5];
  const int*   dst2   = (const int*)d_in[6];
  const float* embedW = (const float*)d_in[7];
  const float* Ws0 = (const float*)d_in[8];
  const float* Wn0 = (const float*)d_in[9];
  const float* b0  = (const float*)d_in[10];
  const float* Ws1 = (const float*)d_in[11];
  const float* Wn1 = (const float*)d_in[12];
  const float* b1  = (const float*)d_in[13];
  const float* Ws2 = (const float*)d_in[14];
  const float* Wn2 = (const float*)d_in[15];
  const float* b2  = (const float*)d_in[16];
  const float* fcW = (const float*)d_in[17];
  const float* fcb = (const float*)d_in[18];
  float* out = (float*)d_out;

  const int NSRC0 = 200000, NDST0 = 50000, NE0 = 800000;
  const int NDST1 = 12500,  NE1 = 200000;
  const int NDST2 = 4096,   NE2 = 65536;

  float* ws = (float*)d_ws;
  size_t off = 0;
  float* H0 = ws + off; off += (size_t)NSRC0 * NFEAT;   // 200000x128
  float* H1 = ws + off; off += (size_t)NDST0 * NFEAT;   // 50000x128
  float* H2 = ws + off; off += (size_t)NDST1 * NFEAT;   // 12500x128
  float* H3 = ws + off; off += (size_t)NDST2 * NFEAT;   // 4096x128
  float* agg0 = ws + off; off += (size_t)NDST0 * NFEAT;
  float* deg0 = ws + off; off += NDST0;
  float* agg1 = ws + off; off += (size_t)NDST1 * NFEAT;
  float* deg1 = ws + off; off += NDST1;
  float* agg2 = ws + off; off += (size_t)NDST2 * NFEAT;
  float* deg2 = ws + off; off += NDST2;

  dim3 blk(256);

  // 1) embed: H0 = x @ embed_W^T  (M=200000, K=512, N=128)
  gemm_nt_wmma<<<NSRC0 / 16, blk, 0, stream>>>(x, embedW, nullptr, H0,
                                               NSRC0, NFEAT, 512, 0);
  // layer 0
  hipMemsetAsync(agg0, 0, sizeof(float) * ((size_t)NDST0 * NFEAT + NDST0), stream);
  sage_scatter<<<(NE0 * 32 + 255) / 256, blk, 0, stream>>>(H0, src0, dst0,
                                                           agg0, deg0, NE0);
  sage_combine_wmma<<<(NDST0 + 15) / 16, blk, 0, stream>>>(H0, agg0, deg0, Ws0,
                                                           Wn0, b0, H1, NDST0, 1);
  // layer 1
  hipMemsetAsync(agg1, 0, sizeof(float) * ((size_t)NDST1 * NFEAT + NDST1), stream);
  sage_scatter<<<(NE1 * 32 + 255) / 256, blk, 0, stream>>>(H1, src1, dst1,
                                                           agg1, deg1, NE1);
  sage_combine_wmma<<<(NDST1 + 15) / 16, blk, 0, stream>>>(H1, agg1, deg1, Ws1,
                                                           Wn1, b1, H2, NDST1, 1);
  // layer 2 (no relu)
  hipMemsetAsync(agg2, 0, sizeof(float) * ((size_t)NDST2 * NFEAT + NDST2), stream);
  sage_scatter<<<(NE2 * 32 + 255) / 256, blk, 0, stream>>>(H2, src2, dst2,
                                                           agg2, deg2, NE2);
  sage_combine_wmma<<<(NDST2 + 15) / 16, blk, 0, stream>>>(H2, agg2, deg2, Ws2,
                                                           Wn2, b2, H3, NDST2, 0);
  // fc: out = H3 @ fc_W^T + fc_b  (M=4096, K=N=128)
  gemm_nt_wmma<<<(NDST2 + 15) / 16, blk, 0, stream>>>(H3, fcW, fcb, out,
                                                      NDST2, NFEAT, NFEAT, 0);
}